// TrainableTokensLayer_25314537242942
// MI455X (gfx1250) — compile-verified
//
#include <hip/hip_runtime.h>
#include <stdint.h>

#define EMBED      2048
#define N_TRAIN    256
#define R          4            // rows staged per block
#define TPB        256
#define ROW_BYTES  (EMBED * 4)  // 8192
#define HALF_BYTES (ROW_BYTES / 2)

typedef float vfloat4 __attribute__((ext_vector_type(4)));

// Fused "trainable tokens" embedding gather:
//   out[row,:] = W[x[row],:] + sum_{n : token_indices[n]==x[row]} delta[n,:]
// delta[n,e] = delta_values[e*N_TRAIN + n]  (COO transpose layout)
//
// Bulk path is pure data movement: global -> LDS -> global via CDNA5 async
// DMA ops (ASYNCcnt), NT store hint so the 128MB write-once output doesn't
// rinse L2 (L2 instead keeps hot W rows for duplicate tokens).
__global__ __launch_bounds__(TPB) void tt_embed_fused(
    const int*   __restrict__ x,
    const float* __restrict__ W,
    const float* __restrict__ delta_values,
    const int*   __restrict__ token_indices,
    float*       __restrict__ out)
{
    __shared__ float smem[R * EMBED];       // 32 KiB staging
    const int tid     = (int)threadIdx.x;
    const int baseRow = (int)blockIdx.x * R;

    // flat shared-aperture address truncates to the LDS byte offset
    const unsigned ldsBase =
        (unsigned)(uintptr_t)(&smem[0]) + (unsigned)(tid * 16);

    // tokens for the R rows this block produces (block-uniform -> scalar loads)
    int tok[R];
    #pragma unroll
    for (int r = 0; r < R; ++r) tok[r] = x[baseRow + r];

    // Stage R gathered W rows into LDS with async global->LDS b128 loads.
    // Each thread owns 2 disjoint 16B chunks per row -> no barriers needed.
    #pragma unroll
    for (int r = 0; r < R; ++r) {
        const unsigned g0 = (unsigned)tok[r] * (unsigned)ROW_BYTES
                          + (unsigned)(tid * 16);          // < 2^31
        const unsigned l0 = ldsBase + (unsigned)(r * ROW_BYTES);
        asm volatile("global_load_async_to_lds_b128 %0, %1, %2"
                     :: "v"(l0), "v"(g0), "s"(W) : "memory");
        asm volatile("global_load_async_to_lds_b128 %0, %1, %2"
                     :: "v"(l0 + HALF_BYTES), "v"(g0 + HALF_BYTES), "s"(W)
                     : "memory");
    }

    #pragma unroll
    for (int r = 0; r < R; ++r) {
        // Loads return done in order; earlier async stores only make this
        // conservative (counted: first 2r+2 loads are guaranteed complete).
        asm volatile("s_wait_asynccnt %0" :: "i"(2 * (R - 1 - r)) : "memory");

        const int row   = baseRow + r;
        const int token = tok[r];

        // equal_range over sorted token_indices (duplicates must sum)
        int lo = 0, hi = N_TRAIN;
        while (lo < hi) { int m = (lo + hi) >> 1;
            if (token_indices[m] <  token) lo = m + 1; else hi = m; }
        int lo2 = lo, hi2 = N_TRAIN;
        while (lo2 < hi2) { int m = (lo2 + hi2) >> 1;
            if (token_indices[m] <= token) lo2 = m + 1; else hi2 = m; }

        const unsigned o0 = (unsigned)row * (unsigned)ROW_BYTES
                          + (unsigned)(tid * 16);
        const unsigned l0 = ldsBase + (unsigned)(r * ROW_BYTES);

        if (lo2 == lo) {
            // Common path (~99.8% of rows): LDS -> global async DMA, NT hint.
            asm volatile("global_store_async_from_lds_b128 %0, %1, %2 th:TH_STORE_NT"
                         :: "v"(o0), "v"(l0), "s"(out) : "memory");
            asm volatile("global_store_async_from_lds_b128 %0, %1, %2 th:TH_STORE_NT"
                         :: "v"(o0 + HALF_BYTES), "v"(l0 + HALF_BYTES), "s"(out)
                         : "memory");
        } else {
            // Rare trained-token row: read staged row, add delta, store.
            const int e0 = tid * 4;
            const int e1 = (EMBED / 2) + tid * 4;
            vfloat4 v0 = *(const vfloat4*)&smem[r * EMBED + e0];
            vfloat4 v1 = *(const vfloat4*)&smem[r * EMBED + e1];
            for (int n = lo; n < lo2; ++n) {
                v0.x += delta_values[(e0 + 0) * N_TRAIN + n];
                v0.y += delta_values[(e0 + 1) * N_TRAIN + n];
                v0.z += delta_values[(e0 + 2) * N_TRAIN + n];
                v0.w += delta_values[(e0 + 3) * N_TRAIN + n];
                v1.x += delta_values[(e1 + 0) * N_TRAIN + n];
                v1.y += delta_values[(e1 + 1) * N_TRAIN + n];
                v1.z += delta_values[(e1 + 2) * N_TRAIN + n];
                v1.w += delta_values[(e1 + 3) * N_TRAIN + n];
            }
            float* orow = out + (size_t)row * EMBED;
            *(vfloat4*)(orow + e0) = v0;
            *(vfloat4*)(orow + e1) = v1;
        }
    }
    asm volatile("s_wait_asynccnt 0" ::: "memory");
}

extern "C" void kernel_launch(void* const* d_in, const int* in_sizes, int n_in,
                              void* d_out, int out_size, void* d_ws, size_t ws_size,
                              hipStream_t stream) {
    const int*   x             = (const int*)  d_in[0];
    const float* W             = (const float*)d_in[1];
    const float* delta_values  = (const float*)d_in[2];
    const int*   token_indices = (const int*)  d_in[3];
    float*       out           = (float*)      d_out;

    const int nRows  = in_sizes[0];   // BATCH*SEQ = 16384, divisible by R
    const int blocks = nRows / R;
    tt_embed_fused<<<blocks, TPB, 0, stream>>>(x, W, delta_values,
                                               token_indices, out);
}